// AdaptiveSplatPositioning_38431367364862
// MI455X (gfx1250) — compile-verified
//
#include <hip/hip_runtime.h>
#include <math.h>

// Problem constants from the reference: B=2, S=2048, D=512, K=64
#define D_DIM    512
#define K_SPLATS 64
#define BT       4096   // B*S tokens

typedef __attribute__((ext_vector_type(2))) float v2f;
typedef __attribute__((ext_vector_type(8))) float v8f;

// ---------------------------------------------------------------------------
// Prep: per-splat ||p||^2, 0.5/clip(|scale|)^2, |importance|  -> d_ws[0..191]
// ---------------------------------------------------------------------------
__global__ __launch_bounds__(64)
void splat_prep_kernel(const float* __restrict__ P,
                       const float* __restrict__ scales,
                       const float* __restrict__ importance,
                       float* __restrict__ ws) {
    int k = threadIdx.x;
    if (k >= K_SPLATS) return;
    float s = fabsf(scales[k]);
    s = fmaxf(s, 1e-6f);
    float inv2s2 = 0.5f / (s * s);
    float im = fabsf(importance[k]);
    const float* row = P + (size_t)k * D_DIM;
    float pn = 0.0f;
    for (int d = 0; d < D_DIM; ++d) {
        float v = row[d];
        pn = fmaf(v, v, pn);
    }
    ws[k]                = pn;       // ||p_k||^2
    ws[K_SPLATS + k]     = inv2s2;   // 0.5 / s^2
    ws[2 * K_SPLATS + k] = im;       // |importance|
}

// ---------------------------------------------------------------------------
// Main: one wave32 per block, 16 tokens x 64 splats per wave.
// dot(x, p) via V_WMMA_F32_16X16X4_F32 accumulating over D=512 (128 k-steps).
// influences = |imp| * exp(-(||x||^2 + ||p||^2 - 2*dot) * 0.5/s^2)
// ---------------------------------------------------------------------------
__global__ __launch_bounds__(32)
void splat_wmma_kernel(const float* __restrict__ X,   // [BT, D]
                       const float* __restrict__ P,   // [K, D]
                       const float* __restrict__ ws,  // [3*K] precomputed
                       float* __restrict__ out) {     // [BT, K]
    const int lane = threadIdx.x;       // 0..31, all lanes active (EXEC = ~0)
    const int half = lane >> 4;         // 0: K=0,1 ; 1: K=2,3 (A/B frag halves)
    const int r    = lane & 15;         // row (A) / col (B) within 16
    const int m0   = blockIdx.x << 4;   // first token of this 16-row tile

    // f32 16x16 C/D accumulators: lanes 0-15 -> M=v, lanes 16-31 -> M=v+8, N=r
    v8f acc0 = {0,0,0,0,0,0,0,0};
    v8f acc1 = {0,0,0,0,0,0,0,0};
    v8f acc2 = {0,0,0,0,0,0,0,0};
    v8f acc3 = {0,0,0,0,0,0,0,0};

    const float* xrow = X + (size_t)(m0 + r) * D_DIM + 2 * half;
    const float* prow = P + (size_t)r * D_DIM + 2 * half;

    float xn = 0.0f;  // partial ||x_row||^2 (this lane's half of the columns)

    for (int d0 = 0; d0 < D_DIM; d0 += 4) {
        // Lead the streaming X reads into cache.
        __builtin_prefetch(xrow + d0 + 256, 0, 0);

        // A fragment: 16x4 tile of X, float2 per lane (contiguous -> b64 load)
        v2f a;
        a.x = xrow[d0];
        a.y = xrow[d0 + 1];
        xn = fmaf(a.x, a.x, xn);
        xn = fmaf(a.y, a.y, xn);

        // B fragments: 4x16 tiles of P^T for the four 16-splat column groups
        v2f b0, b1, b2, b3;
        b0.x = prow[d0 + 0 * 16 * D_DIM]; b0.y = prow[d0 + 0 * 16 * D_DIM + 1];
        b1.x = prow[d0 + 1 * 16 * D_DIM]; b1.y = prow[d0 + 1 * 16 * D_DIM + 1];
        b2.x = prow[d0 + 2 * 16 * D_DIM]; b2.y = prow[d0 + 2 * 16 * D_DIM + 1];
        b3.x = prow[d0 + 3 * 16 * D_DIM]; b3.y = prow[d0 + 3 * 16 * D_DIM + 1];

        acc0 = __builtin_amdgcn_wmma_f32_16x16x4_f32(false, a, false, b0,
                                                     (short)0, acc0, false, false);
        acc1 = __builtin_amdgcn_wmma_f32_16x16x4_f32(false, a, false, b1,
                                                     (short)0, acc1, false, false);
        acc2 = __builtin_amdgcn_wmma_f32_16x16x4_f32(false, a, false, b2,
                                                     (short)0, acc2, false, false);
        acc3 = __builtin_amdgcn_wmma_f32_16x16x4_f32(false, a, false, b3,
                                                     (short)0, acc3, false, false);
    }

    // Combine the two half-wave partial token norms: lane r & lane r+16 each
    // hold half of ||x_{m0+r}||^2.
    float xn_full = xn + __shfl_xor(xn, 16, 32);

    // Broadcast: this lane's output rows are M = v + 8*half, v = 0..7.
    float rown[8];
#pragma unroll
    for (int v = 0; v < 8; ++v)
        rown[v] = __shfl(xn_full, v + 8 * half, 32);

    const float* pnorm  = ws;
    const float* inv2s2 = ws + K_SPLATS;
    const float* impabs = ws + 2 * K_SPLATS;

#pragma unroll
    for (int nt = 0; nt < 4; ++nt) {
        const v8f acc = (nt == 0) ? acc0 : (nt == 1) ? acc1
                      : (nt == 2) ? acc2 : acc3;
        const int col = nt * 16 + r;
        const float pn = pnorm[col];
        const float is = inv2s2[col];
        const float im = impabs[col];
#pragma unroll
        for (int v = 0; v < 8; ++v) {
            const int row = v + 8 * half;
            float d2 = rown[v] + pn - 2.0f * acc[v];
            d2 = fmaxf(d2, 0.0f);
            out[(size_t)(m0 + row) * K_SPLATS + col] = im * __expf(-d2 * is);
        }
    }
}

extern "C" void kernel_launch(void* const* d_in, const int* in_sizes, int n_in,
                              void* d_out, int out_size, void* d_ws, size_t ws_size,
                              hipStream_t stream) {
    const float* X      = (const float*)d_in[0];  // token_embeddings [2,2048,512]
    const float* P      = (const float*)d_in[1];  // splat_positions  [64,512]
    const float* scales = (const float*)d_in[2];  // splat_scales     [64]
    const float* imp    = (const float*)d_in[3];  // splat_importance [64]
    float* out = (float*)d_out;                   // [2,2048,64]
    float* ws  = (float*)d_ws;                    // needs 3*64 floats = 768 B

    splat_prep_kernel<<<1, 64, 0, stream>>>(P, scales, imp, ws);
    splat_wmma_kernel<<<BT / 16, 32, 0, stream>>>(X, P, ws, out);
}